// SS2D_72713796321848
// MI455X (gfx1250) — compile-verified
//
#include <hip/hip_runtime.h>
#include <hip/hip_bf16.h>
#include <math.h>

// ---------------------------------------------------------------------------
// SS2D (Mamba-style 2D selective scan), fp32, MI455X / gfx1250.
// Pipeline of 7 kernels; GEMMs use V_WMMA_F32_16X16X4_F32 (native fp32 WMMA).
// Shapes: B=32, H=W=32, L=1024, DM=128, DIN=64, N=16, R=8.
// ---------------------------------------------------------------------------

#define BB   32
#define HH   32
#define WW   32
#define LL   1024       // H*W
#define DM   128
#define DIN  64
#define NS   16
#define RR   8
#define CPROJ (RR + 2*NS)   // 40

typedef float v2f __attribute__((ext_vector_type(2)));
typedef float v8f __attribute__((ext_vector_type(8)));

__device__ __forceinline__ float silu_f(float v) {
    return v * (1.0f / (1.0f + __expf(-v)));
}
__device__ __forceinline__ float softplus_f(float v) {
    return (v > 20.0f) ? v : log1pf(expf(v));
}

// ---------------------------------------------------------------------------
// K1: in-proj GEMM  xz[m,e] = sum_c x[m,c] * w_in[e,c]   (M=32768,N=128,K=128)
// cols 0..63  -> xc_t stored transposed (B, DIN, H, W) for the conv
// cols 64..127-> z   stored (B, L, DIN) for the gate
// One wave = one 16x16 C tile; block = 8 waves covering all N=128 for one M tile.
// ---------------------------------------------------------------------------
__global__ __launch_bounds__(256) void k_in_proj(
    const float* __restrict__ x, const float* __restrict__ w_in,
    float* __restrict__ xc_t, float* __restrict__ z)
{
    const int wave = threadIdx.x >> 5;
    const int lane = threadIdx.x & 31;
    const int lo16 = lane & 15;
    const int half = lane >> 4;          // 0: K={0,1}, 1: K={2,3}
    const int m0   = blockIdx.x << 4;    // grid.x = 32768/16 = 2048
    const int n0   = wave << 4;          // 8 waves x 16 = 128 columns

    const int arow = m0 + lo16;          // A row (per lane)
    const int bcol = n0 + lo16;          // B col (per lane)
    const int kh   = half << 1;          // k offset within 4-chunk

    v8f c = {};
    #pragma unroll
    for (int kk = 0; kk < DM; kk += 4) {
        v2f a = *(const v2f*)(x    + arow * DM + kk + kh);   // A[M=arow, K=kk+kh..+1]
        v2f b = *(const v2f*)(w_in + bcol * DM + kk + kh);   // B[K][N]=w_in[N,K]
        c = __builtin_amdgcn_wmma_f32_16x16x4_f32(
                false, a, false, b, (short)0, c, false, false);
    }

    #pragma unroll
    for (int i = 0; i < 8; ++i) {
        const int m = m0 + i + (half << 3);
        const int e = bcol;
        const float v = c[i];
        const int b_ = m >> 10;          // batch
        const int l  = m & (LL - 1);     // pixel
        if (e < DIN) {                   // uniform per wave (n0 multiple of 16)
            xc_t[((b_ * DIN + e) << 10) + l] = v;
        } else {
            z[(m << 6) + (e - DIN)] = v;
        }
    }
}

// ---------------------------------------------------------------------------
// K2: depthwise 3x3 conv + bias + SiLU  ->  u (B, DIN, L)
// ---------------------------------------------------------------------------
__global__ __launch_bounds__(256) void k_dwconv(
    const float* __restrict__ xc_t, const float* __restrict__ conv_w,
    const float* __restrict__ conv_b, float* __restrict__ u)
{
    const int idx = blockIdx.x * blockDim.x + threadIdx.x;  // B*DIN*L
    const int l  = idx & (LL - 1);
    const int bd = idx >> 10;
    const int d  = bd & (DIN - 1);
    const int hy = l >> 5, wx = l & 31;
    const float* __restrict__ src = xc_t + (bd << 10);
    const float* __restrict__ wk  = conv_w + d * 9;
    float acc = conv_b[d];
    #pragma unroll
    for (int dy = -1; dy <= 1; ++dy) {
        #pragma unroll
        for (int dx = -1; dx <= 1; ++dx) {
            const int yy = hy + dy, xx = wx + dx;
            if ((unsigned)yy < HH && (unsigned)xx < WW)
                acc = fmaf(wk[(dy + 1) * 3 + (dx + 1)], src[yy * WW + xx], acc);
        }
    }
    u[idx] = silu_f(acc);
}

// ---------------------------------------------------------------------------
// K3: x-proj GEMM  x_dbl[b,c,l] = sum_d xproj_w[c,d] * u[b,d,l]
// Per batch: (40 x 64) x (64 x 1024). A = xproj (shared), B = u slice.
// grid = (8 l-groups, 3 c-tiles, 32 batches); block = 8 waves (one l-tile each).
// ---------------------------------------------------------------------------
__global__ __launch_bounds__(256) void k_xproj(
    const float* __restrict__ u, const float* __restrict__ xproj_w,
    float* __restrict__ x_dbl)
{
    const int wave = threadIdx.x >> 5;
    const int lane = threadIdx.x & 31;
    const int lo16 = lane & 15;
    const int half = lane >> 4;
    const int kh   = half << 1;
    const int l0   = ((blockIdx.x << 3) + wave) << 4;   // 64 l-tiles
    const int c0   = blockIdx.y << 4;                   // 3 c-tiles (40 rows)
    const int b_   = blockIdx.z;

    const int crow  = c0 + lo16;
    const int crowc = crow < CPROJ ? crow : (CPROJ - 1); // clamp (branchless)
    const float cmask = (crow < CPROJ) ? 1.0f : 0.0f;

    const float* __restrict__ ub = u + ((b_ * DIN) << 10) + l0 + lo16;

    v8f c = {};
    #pragma unroll
    for (int kk = 0; kk < DIN; kk += 4) {
        v2f a;
        a.x = xproj_w[crowc * DIN + kk + kh]     * cmask;
        a.y = xproj_w[crowc * DIN + kk + kh + 1] * cmask;
        v2f b;
        b.x = ub[(kk + kh)     << 10];           // u[b, kk+kh,   l0+lo16]
        b.y = ub[(kk + kh + 1) << 10];
        c = __builtin_amdgcn_wmma_f32_16x16x4_f32(
                false, a, false, b, (short)0, c, false, false);
    }

    #pragma unroll
    for (int i = 0; i < 8; ++i) {
        const int cr = c0 + i + (half << 3);
        if (cr < CPROJ)
            x_dbl[((b_ * CPROJ + cr) << 10) + l0 + lo16] = c[i];
    }
}

// ---------------------------------------------------------------------------
// K4: delta[b,d,l] = softplus( sum_r dts[b,r,l]*dtproj_w[d,r] + dtproj_b[d] )
// (K=8 is too small to reward WMMA; pure VALU, 33 MFLOP total.)
// ---------------------------------------------------------------------------
__global__ __launch_bounds__(256) void k_delta(
    const float* __restrict__ x_dbl, const float* __restrict__ dtproj_w,
    const float* __restrict__ dtproj_b, float* __restrict__ delta)
{
    const int idx = blockIdx.x * blockDim.x + threadIdx.x;  // B*DIN*L
    const int l  = idx & (LL - 1);
    const int bd = idx >> 10;
    const int d  = bd & (DIN - 1);
    const int b_ = bd >> 6;
    const float* __restrict__ dts = x_dbl + ((b_ * CPROJ) << 10) + l;
    const float* __restrict__ wr  = dtproj_w + d * RR;
    float acc = dtproj_b[d];
    #pragma unroll
    for (int r = 0; r < RR; ++r)
        acc = fmaf(wr[r], dts[r << 10], acc);
    delta[idx] = softplus_f(acc);
}

// ---------------------------------------------------------------------------
// K5: selective scan. Wave32 packs TWO (b,d) recurrences: lane = 16*pair + n.
// h_n updated serially over l=0..1023; y_t reduced over n via shfl_xor within
// each 16-lane half. 1024 waves total -> good WGP occupancy on the serial path.
// y stored (B, L, DIN) for the LN/out-proj that follow.
// ---------------------------------------------------------------------------
__global__ __launch_bounds__(256) void k_scan(
    const float* __restrict__ u, const float* __restrict__ x_dbl,
    const float* __restrict__ delta, const float* __restrict__ A_logs,
    const float* __restrict__ Ds, float* __restrict__ y)
{
    const int lane = threadIdx.x & 31;
    const int wv   = blockIdx.x * (blockDim.x >> 5) + (threadIdx.x >> 5);
    const int pair = lane >> 4;
    const int n    = lane & 15;
    const int bd   = (wv << 1) + pair;        // (b,d) in [0, 2048)
    const int b_   = bd >> 6;
    const int d    = bd & (DIN - 1);

    const float Adn = -__expf(A_logs[d * NS + n]);
    const float Dd  = Ds[d];
    const float* __restrict__ Bs = x_dbl + ((b_ * CPROJ + RR      + n) << 10);
    const float* __restrict__ Cs = x_dbl + ((b_ * CPROJ + RR + NS + n) << 10);
    const float* __restrict__ up = u     + (bd << 10);
    const float* __restrict__ dp = delta + (bd << 10);
    float* __restrict__ yp = y + ((b_ << 10) << 6) + d;

    float h = 0.0f;
    for (int l = 0; l < LL; ++l) {
        const float dl  = dp[l];
        const float ul  = up[l];
        const float dA  = __expf(dl * Adn);
        const float dBu = dl * ul * Bs[l];
        h = fmaf(dA, h, dBu);
        float yv = h * Cs[l];
        yv += __shfl_xor(yv, 8, 32);   // stays inside each 16-lane half
        yv += __shfl_xor(yv, 4, 32);
        yv += __shfl_xor(yv, 2, 32);
        yv += __shfl_xor(yv, 1, 32);
        if (n == 0) yp[l << 6] = fmaf(Dd, ul, yv);
    }
}

// ---------------------------------------------------------------------------
// K6: LayerNorm over DIN=64 + silu(z) gate. One wave per (b,l) row.
// ---------------------------------------------------------------------------
__global__ __launch_bounds__(256) void k_ln_gate(
    const float* __restrict__ y, const float* __restrict__ z,
    const float* __restrict__ ln_g, const float* __restrict__ ln_b,
    float* __restrict__ yg)
{
    const int lane = threadIdx.x & 31;
    const int row  = blockIdx.x * (blockDim.x >> 5) + (threadIdx.x >> 5);
    const float* __restrict__ yr = y + (row << 6);
    const float* __restrict__ zr = z + (row << 6);
    const float v0 = yr[lane], v1 = yr[lane + 32];
    float s  = v0 + v1;
    float sq = v0 * v0 + v1 * v1;
    #pragma unroll
    for (int off = 16; off; off >>= 1) {
        s  += __shfl_xor(s,  off, 32);
        sq += __shfl_xor(sq, off, 32);
    }
    const float mu  = s * (1.0f / DIN);
    const float var = sq * (1.0f / DIN) - mu * mu;
    const float rs  = rsqrtf(var + 1e-5f);
    yg[(row << 6) + lane]      = fmaf((v0 - mu) * rs, ln_g[lane],      ln_b[lane])      * silu_f(zr[lane]);
    yg[(row << 6) + lane + 32] = fmaf((v1 - mu) * rs, ln_g[lane + 32], ln_b[lane + 32]) * silu_f(zr[lane + 32]);
}

// ---------------------------------------------------------------------------
// K7: out-proj GEMM  out[m,c] = sum_d yg[m,d] * w_out[c,d]  (M=32768,N=128,K=64)
// ---------------------------------------------------------------------------
__global__ __launch_bounds__(256) void k_out_proj(
    const float* __restrict__ yg, const float* __restrict__ w_out,
    float* __restrict__ out)
{
    const int wave = threadIdx.x >> 5;
    const int lane = threadIdx.x & 31;
    const int lo16 = lane & 15;
    const int half = lane >> 4;
    const int kh   = half << 1;
    const int m0   = blockIdx.x << 4;    // 2048 M tiles
    const int n0   = wave << 4;          // 8 waves x 16 = 128 columns

    const int arow = m0 + lo16;
    const int bcol = n0 + lo16;

    v8f c = {};
    #pragma unroll
    for (int kk = 0; kk < DIN; kk += 4) {
        v2f a = *(const v2f*)(yg    + arow * DIN + kk + kh);
        v2f b = *(const v2f*)(w_out + bcol * DIN + kk + kh);
        c = __builtin_amdgcn_wmma_f32_16x16x4_f32(
                false, a, false, b, (short)0, c, false, false);
    }

    #pragma unroll
    for (int i = 0; i < 8; ++i) {
        const int m = m0 + i + (half << 3);
        out[m * DM + bcol] = c[i];
    }
}

// ---------------------------------------------------------------------------
// Launch
// ---------------------------------------------------------------------------
extern "C" void kernel_launch(void* const* d_in, const int* in_sizes, int n_in,
                              void* d_out, int out_size, void* d_ws, size_t ws_size,
                              hipStream_t stream) {
    const float* x        = (const float*)d_in[0];
    const float* w_in     = (const float*)d_in[1];
    const float* conv_w   = (const float*)d_in[2];
    const float* conv_b   = (const float*)d_in[3];
    const float* xproj_w  = (const float*)d_in[4];
    const float* dtproj_w = (const float*)d_in[5];
    const float* dtproj_b = (const float*)d_in[6];
    const float* A_logs   = (const float*)d_in[7];
    const float* Ds       = (const float*)d_in[8];
    const float* ln_g     = (const float*)d_in[9];
    const float* ln_b     = (const float*)d_in[10];
    const float* w_out    = (const float*)d_in[11];
    float* out = (float*)d_out;

    // Workspace layout (floats). Total ~13.9M floats (~56 MB) -> fits 192MB L2.
    float* ws = (float*)d_ws;
    float* xc_t  = ws;                         // B*DIN*L = 2097152
    float* z     = xc_t  + BB * DIN * LL;      // B*L*DIN = 2097152
    float* u     = z     + BB * LL * DIN;      // B*DIN*L = 2097152
    float* x_dbl = u     + BB * DIN * LL;      // B*40*L  = 1310720
    float* delta = x_dbl + BB * CPROJ * LL;    // B*DIN*L = 2097152
    float* y     = delta + BB * DIN * LL;      // B*L*DIN = 2097152
    float* yg    = y     + BB * LL * DIN;      // B*L*DIN = 2097152

    const int M = BB * LL;                     // 32768

    // K1: in-proj (WMMA f32)
    k_in_proj<<<M / 16, 256, 0, stream>>>(x, w_in, xc_t, z);
    // K2: depthwise conv + SiLU
    k_dwconv<<<(BB * DIN * LL) / 256, 256, 0, stream>>>(xc_t, conv_w, conv_b, u);
    // K3: x-proj (WMMA f32)
    k_xproj<<<dim3(8, 3, BB), 256, 0, stream>>>(u, xproj_w, x_dbl);
    // K4: delta
    k_delta<<<(BB * DIN * LL) / 256, 256, 0, stream>>>(x_dbl, dtproj_w, dtproj_b, delta);
    // K5: selective scan (2 recurrences per wave, 1024 waves)
    k_scan<<<(BB * DIN / 2) / 8, 256, 0, stream>>>(u, x_dbl, delta, A_logs, Ds, y);
    // K6: LayerNorm + gate
    k_ln_gate<<<M / 8, 256, 0, stream>>>(y, z, ln_g, ln_b, yg);
    // K7: out-proj (WMMA f32)
    k_out_proj<<<M / 16, 256, 0, stream>>>(yg, w_out, out);
}